// AccumulatingModule_83099027243635
// MI455X (gfx1250) — compile-verified
//
#include <hip/hip_runtime.h>

// ---- problem constants (from reference) ----
#define NUM_QT   65
#define NUM_OT   151
#define PAIR_NUM 90
#define BATCH    8192
#define BOX      10
#define OT2      (NUM_OT * NUM_OT)   // 22801

typedef __attribute__((ext_vector_type(2))) float v2f;
typedef __attribute__((ext_vector_type(4))) float v4f;
typedef __attribute__((ext_vector_type(8))) float v8f;

// --------------------------------------------------------------------------
// Kernel 1: streaming copy of score_matrix (533 MB) into d_out.
// B128 vector loads/stores, non-temporal (output >> 192MB L2).
// --------------------------------------------------------------------------
__global__ __launch_bounds__(256) void copy_score_kernel(
    const float* __restrict__ src, float* __restrict__ dst, long n4, long n) {
  const long tid    = (long)blockIdx.x * blockDim.x + threadIdx.x;
  const long stride = (long)gridDim.x * blockDim.x;

  const v4f* __restrict__ s4 = (const v4f*)src;
  v4f* __restrict__       d4 = (v4f*)dst;
  for (long k = tid; k < n4; k += stride) {
    v4f v = __builtin_nontemporal_load(&s4[k]);
    __builtin_nontemporal_store(v, &d4[k]);
  }
  // scalar tail (out_size % 4 == 2)
  for (long k = n4 * 4 + tid; k < n; k += stride) {
    dst[k] = src[k];
  }
}

// --------------------------------------------------------------------------
// Kernel 2: one wave32 per batch.
//   D = A x B with V_WMMA_F32_16X16X4_F32, where only K=0 is nonzero:
//     A[m,0] = att[m], B[0,n] = att[n]  ->  D[m,n] = att[m]*att[n]
//   A 16x4 f32 layout: lanes 0-15 VGPR0=K0, VGPR1=K1; lanes 16-31 K2,K3.
//   B/D: row striped across lanes within a VGPR (D: VGPR r -> M=r / M=r+8).
//   Each lane then scatter-adds its valid off-diagonal products.
// --------------------------------------------------------------------------
__global__ __launch_bounds__(256) void scatter_wmma_kernel(
    const int* __restrict__ obj_label,   // [BATCH, BOX]
    const int* __restrict__ qus_type,    // [BATCH]
    const float* __restrict__ attention, // [BATCH, BOX]
    float* __restrict__ out)             // [NUM_QT, PAIR_NUM, NUM_OT, NUM_OT]
{
  const int lane = threadIdx.x & 31;
  const int b    = blockIdx.x * (blockDim.x >> 5) + (threadIdx.x >> 5);
  if (b >= BATCH) return;  // wave-uniform; EXEC stays all-1s inside

  const int*   lab = obj_label + b * BOX;
  const float* att = attention + b * BOX;
  const int    qt  = qus_type[b];

  const int n    = lane & 15;   // column index N in the 16x16 tile
  const int half = lane >> 4;   // D: VGPR r holds M=r (half=0) / M=r+8 (half=1)

  // Operand: att[lane] into the K=0 slot, everything else zero.
  // (lanes >= 10 and the K=1..3 VGPR/lane slots are zeroed -> exact rank-1.)
  float a0 = (lane < BOX) ? att[lane] : 0.0f;
  v2f a;
  a.x = a0;
  a.y = 0.0f;

  v8f c = {};
  // D[m,n] = att[m] * att[n]
  v8f d = __builtin_amdgcn_wmma_f32_16x16x4_f32(
      /*neg_a=*/false, a, /*neg_b=*/false, a,
      /*c_mod=*/(short)0, c, /*reuse_a=*/false, /*reuse_b=*/false);

  const int ol1 = (n < BOX) ? lab[n] : 0;  // obj_label[b, j=n]
  float* const qbase = out + (size_t)qt * PAIR_NUM * OT2;

  #pragma unroll
  for (int r = 0; r < 8; ++r) {
    const int m = r + half * 8;  // row index M (i)
    if (m < BOX && n < BOX && m != n) {
      // pair index p for (i=m, j=n): row-major off-diagonal enumeration
      const int p   = m * (BOX - 1) + (n < m ? n : n - 1);
      const int ol2 = lab[m];    // obj_label[b, i=m]
      unsafeAtomicAdd(qbase + (size_t)p * OT2 + ol1 * NUM_OT + ol2, d[r]);
    }
  }
}

// --------------------------------------------------------------------------
// Launcher: copy then scatter (same stream -> ordered).
// --------------------------------------------------------------------------
extern "C" void kernel_launch(void* const* d_in, const int* in_sizes, int n_in,
                              void* d_out, int out_size, void* d_ws, size_t ws_size,
                              hipStream_t stream) {
  const int*   obj_label = (const int*)d_in[0];    // [8192,10] int32
  const int*   qus_type  = (const int*)d_in[1];    // [8192]    int32
  const float* attention = (const float*)d_in[2];  // [8192,10] f32
  const float* score     = (const float*)d_in[3];  // [65,90,151,151] f32
  float*       out       = (float*)d_out;

  const long N  = (long)out_size;   // 133,385,850
  const long n4 = N >> 2;

  copy_score_kernel<<<4096, 256, 0, stream>>>(score, out, n4, N);

  // 8 waves per block, one wave per batch
  const int blocks = BATCH / 8;     // 1024
  scatter_wmma_kernel<<<blocks, 256, 0, stream>>>(obj_label, qus_type, attention, out);
}